// RayleighKLLoss_mat2_41790031790570
// MI455X (gfx1250) — compile-verified
//
#include <hip/hip_runtime.h>
#include <math.h>

#define BINS      50
#define EPSF      1e-8f
#define CLAMP_MIN 1e-6f
#define BATCH     32
#define HW        (512 * 512)        /* pixels per sample            */
#define SSTRIDE   (2 * HW)           /* floats per sample (2 chans)  */
#define QPS       (HW / 4)           /* float4 quads per sample = 65536 */
#define BLK_PER_S 32                 /* hist blocks per sample        */
#define CHUNK     (HW / BLK_PER_S)   /* 8192 pixels per hist block    */

typedef __attribute__((ext_vector_type(2))) float v2f;
typedef __attribute__((ext_vector_type(8))) float v8f;

// Workspace layout (uint32 words):
//   ws[0] = global min (float bits, all values > 0 so u32 order == float order)
//   ws[1] = global max (float bits)
//   ws[2 .. 2 + 2*BATCH*BINS) = histogram counts: [input][sample][bin]

__global__ void k_init(unsigned int* __restrict__ ws) {
    int t = threadIdx.x + blockIdx.x * blockDim.x;
    if (t == 0) { ws[0] = 0x7F800000u; ws[1] = 0u; }   // +inf, 0
    for (int i = t; i < 2 * BATCH * BINS; i += blockDim.x * gridDim.x)
        ws[2 + i] = 0u;
}

__device__ __forceinline__ float mag1(float a, float c) {
    return fmaxf(sqrtf(a * a + c * c), CLAMP_MIN);
}

__global__ void k_minmax(const float* __restrict__ pred,
                         const float* __restrict__ tru,
                         unsigned int* __restrict__ ws) {
    const int QTOT = 2 * BATCH * QPS;          // 2^22 quads total
    float vmin = INFINITY, vmax = 0.0f;
    const int stride = blockDim.x * gridDim.x;
    for (int q = threadIdx.x + blockIdx.x * blockDim.x; q < QTOT; q += stride) {
        const int i = q >> 21;                 // BATCH*QPS = 2^21
        const int r = q & ((1 << 21) - 1);
        const int b = r >> 16;                 // QPS = 2^16
        const int p = (r & 0xFFFF) * 4;
        const float* src  = (i == 0) ? pred : tru;
        const float* base = src + (size_t)b * SSTRIDE;
        const float4 a = *(const float4*)(base + p);
        const float4 c = *(const float4*)(base + HW + p);
        float m0 = mag1(a.x, c.x), m1 = mag1(a.y, c.y);
        float m2 = mag1(a.z, c.z), m3 = mag1(a.w, c.w);
        vmin = fminf(vmin, fminf(fminf(m0, m1), fminf(m2, m3)));
        vmax = fmaxf(vmax, fmaxf(fmaxf(m0, m1), fmaxf(m2, m3)));
    }
    // wave32 butterfly reduction
    for (int off = 16; off; off >>= 1) {
        vmin = fminf(vmin, __shfl_xor(vmin, off, 32));
        vmax = fmaxf(vmax, __shfl_xor(vmax, off, 32));
    }
    __shared__ float smin[8], smax[8];
    const int wave = threadIdx.x >> 5, lane = threadIdx.x & 31;
    if (lane == 0) { smin[wave] = vmin; smax[wave] = vmax; }
    __syncthreads();
    if (threadIdx.x == 0) {
        float bmin = smin[0], bmax = smax[0];
        for (int w = 1; w < (int)(blockDim.x >> 5); ++w) {
            bmin = fminf(bmin, smin[w]);
            bmax = fmaxf(bmax, smax[w]);
        }
        atomicMin(&ws[0], __float_as_uint(bmin));
        atomicMax(&ws[1], __float_as_uint(bmax));
    }
}

__global__ void k_hist(const float* __restrict__ pred,
                       const float* __restrict__ tru,
                       unsigned int* __restrict__ ws) {
    __shared__ unsigned int h[8][BINS + 2];    // one sub-hist per wave
    const int wave = threadIdx.x >> 5;
    for (int i = threadIdx.x; i < 8 * (BINS + 2); i += blockDim.x)
        (&h[0][0])[i] = 0u;
    __syncthreads();

    const float mn = __uint_as_float(ws[0]);
    const float mx = __uint_as_float(ws[1]);
    const float scale = (float)BINS / (mx - mn);

    const int z   = blockIdx.x;                // 2048 blocks total
    const int i   = z / (BATCH * BLK_PER_S);
    const int r   = z - i * (BATCH * BLK_PER_S);
    const int b   = r / BLK_PER_S;
    const int blk = r - b * BLK_PER_S;
    const float* src  = (i == 0) ? pred : tru;
    const float* base = src + (size_t)b * SSTRIDE;
    const int p0 = blk * CHUNK;

    for (int t = threadIdx.x; t < CHUNK / 4; t += blockDim.x) {
        const int p = p0 + t * 4;
        if (t + (int)blockDim.x < CHUNK / 4) {  // prefetch next iteration
            __builtin_prefetch(base + p + 4 * blockDim.x, 0, 3);
            __builtin_prefetch(base + HW + p + 4 * blockDim.x, 0, 3);
        }
        const float4 a = *(const float4*)(base + p);
        const float4 c = *(const float4*)(base + HW + p);
        float m[4] = { mag1(a.x, c.x), mag1(a.y, c.y),
                       mag1(a.z, c.z), mag1(a.w, c.w) };
#pragma unroll
        for (int j = 0; j < 4; ++j) {
            int idx = (int)floorf((m[j] - mn) * scale);
            idx = min(max(idx, 0), BINS - 1);
            atomicAdd(&h[wave][idx], 1u);      // ds_add_u32
        }
    }
    __syncthreads();
    unsigned int* gh = ws + 2 + (size_t)(i * BATCH + b) * BINS;
    for (int n = threadIdx.x; n < BINS; n += blockDim.x) {
        unsigned int s = 0;
#pragma unroll
        for (int w = 0; w < 8; ++w) s += h[w][n];
        atomicAdd(&gh[n], s);                  // global_atomic_add_u32
    }
}

// Finalize: WMMA row sums of the (32 x 50) (count+eps) matrices, then KL.
// One wave per (input, 16-row tile): A holds counts per the ISA 16x4 f32
// layout; B is all-ones so D[m][n] = rowsum[m] for every column n.
__global__ void k_finalize(const unsigned int* __restrict__ ws,
                           float* __restrict__ out) {
    __shared__ float sums[2][BATCH];
    __shared__ float red[4];
    const unsigned int* cnt = ws + 2;
    const int lane = threadIdx.x & 31;
    const int wave = threadIdx.x >> 5;         // 4 waves
    const int i       = wave >> 1;             // input (pred/true)
    const int rowbase = (wave & 1) * 16;       // sample tile
    const unsigned int* hbase = cnt + i * (BATCH * BINS);

    const int m    = lane & 15;
    const int krow = (lane < 16) ? 0 : 2;      // ISA 16x4 f32 A layout
    v8f acc = {};
    v2f bones; bones.x = 1.0f; bones.y = 1.0f;
    for (int k0 = 0; k0 < 52; k0 += 4) {       // K padded 50 -> 52
        const int ka = k0 + krow, kb = ka + 1;
        v2f a;
        a.x = (ka < BINS) ? ((float)hbase[(rowbase + m) * BINS + ka] + EPSF) : 0.0f;
        a.y = (kb < BINS) ? ((float)hbase[(rowbase + m) * BINS + kb] + EPSF) : 0.0f;
        acc = __builtin_amdgcn_wmma_f32_16x16x4_f32(
            false, a, false, bones, (short)0, acc, false, false);
    }
    // D layout: lanes 0-15 -> rows rowbase+0..7 in acc[0..7];
    //           lanes 16-31 -> rows rowbase+8..15 in acc[0..7].
    if (lane == 0)
        for (int r = 0; r < 8; ++r) sums[i][rowbase + r] = acc[r];
    if (lane == 16)
        for (int r = 0; r < 8; ++r) sums[i][rowbase + 8 + r] = acc[r];
    __syncthreads();

    float part = 0.0f;
    for (int t = threadIdx.x; t < BATCH * BINS; t += blockDim.x) {
        const int b = t / BINS, n = t - b * BINS;
        const float hp = ((float)cnt[b * BINS + n] + EPSF) / sums[0][b];
        const float ht = ((float)cnt[BATCH * BINS + b * BINS + n] + EPSF) / sums[1][b];
        part += ht * logf(ht / hp);
    }
    for (int off = 16; off; off >>= 1) part += __shfl_xor(part, off, 32);
    if (lane == 0) red[wave] = part;
    __syncthreads();
    if (threadIdx.x == 0)
        out[0] = (red[0] + red[1] + red[2] + red[3]) / (float)BATCH;
}

extern "C" void kernel_launch(void* const* d_in, const int* in_sizes, int n_in,
                              void* d_out, int out_size, void* d_ws, size_t ws_size,
                              hipStream_t stream) {
    const float* pred = (const float*)d_in[0];
    const float* tru  = (const float*)d_in[1];
    unsigned int* ws  = (unsigned int*)d_ws;
    float* out        = (float*)d_out;

    k_init<<<13, 256, 0, stream>>>(ws);
    k_minmax<<<4096, 256, 0, stream>>>(pred, tru, ws);
    k_hist<<<2 * BATCH * BLK_PER_S, 256, 0, stream>>>(pred, tru, ws);
    k_finalize<<<1, 128, 0, stream>>>(ws, out);
}